// SPPPViT_72009421685116
// MI455X (gfx1250) — compile-verified
//
#include <hip/hip_runtime.h>

// Problem constants (fixed by the reference harness)
#define IMG    224
#define PSZ    4
#define RSEG   16
#define BATCH  32
#define DIM    768
#define NPX    (IMG / PSZ)       // 56
#define NPATCH (NPX * NPX)       // 3136
#define DTILES (DIM / 16)        // 48

typedef __attribute__((ext_vector_type(2))) float v2f;
typedef __attribute__((ext_vector_type(8))) float v8f;

// ---------------- kernel 0: zero the B*R count bins ----------------
__global__ void sppp_zero_counts(int* __restrict__ cnt) {
    int t = threadIdx.x;
    if (t < BATCH * RSEG) cnt[t] = 0;
}

// ---------------- kernel 1: per-patch dominant segment + counts ----------------
// One thread per patch. Mode of 16 labels computed entirely in registers
// (ascending label scan with strict '>' => tie-break to smallest label,
// matching the reference argmax). Counts aggregated in LDS, then 32 global
// atomics per block.
__global__ void sppp_seg_mode(const int* __restrict__ seg,
                              int* __restrict__ seg_ids,
                              int* __restrict__ cnt) {
    __shared__ int hist[2 * RSEG];
    int t = threadIdx.x;
    if (t < 2 * RSEG) hist[t] = 0;
    __syncthreads();

    int p  = blockIdx.x * 256 + t;
    int b0 = (blockIdx.x * 256) / NPATCH;   // first batch index touched by block
    if (p < BATCH * NPATCH) {
        int b  = p / NPATCH;
        int n  = p - b * NPATCH;
        int py = n / NPX;
        int px = n - py * NPX;
        const int* base = seg + ((size_t)b * IMG + (size_t)py * PSZ) * IMG + px * PSZ;

        int lab[16];
#pragma unroll
        for (int r = 0; r < PSZ; ++r) {            // 16-byte aligned row loads
            int4 v = *(const int4*)(base + r * IMG);
            lab[r * 4 + 0] = v.x; lab[r * 4 + 1] = v.y;
            lab[r * 4 + 2] = v.z; lab[r * 4 + 3] = v.w;
        }
        int best = 0, bestc = -1;
#pragma unroll
        for (int r = 0; r < RSEG; ++r) {
            int c = 0;
#pragma unroll
            for (int i = 0; i < 16; ++i) c += (lab[i] == r) ? 1 : 0;
            if (c > bestc) { bestc = c; best = r; }  // ties keep smallest r
        }
        seg_ids[p] = best;
        atomicAdd(&hist[(b - b0) * RSEG + best], 1);
    }
    __syncthreads();
    if (t < 2 * RSEG) {
        int v  = hist[t];
        int bb = b0 + (t >> 4);
        if (v != 0 && bb < BATCH) atomicAdd(&cnt[bb * RSEG + (t & 15)], v);
    }
}

// ---------------- kernel 2: one-hot WMMA pooling ----------------
// Block = 256 threads = 8 waves. Block owns (batch b, 16-wide D tile).
// Each wave accumulates a 16(R) x 16(Dcols) f32 tile over its 392-patch
// K range with v_wmma_f32_16x16x4_f32; A is the one-hot segment matrix.
// 32-bit WMMA operand layout (ISA 7.12.2):
//   A 16x4 : lane M = lane&15 ; VGPR0 K = 2*(lane>>4), VGPR1 K = 2*(lane>>4)+1
//   B 4x16 : lane N = lane&15 ; same K split
//   C/D    : VGPR v : M = v + 8*(lane>>4), N = lane&15
__global__ void sppp_pool_wmma(const float* __restrict__ emb,
                               const int* __restrict__ seg_ids,
                               const int* __restrict__ cnt,
                               float* __restrict__ out) {
    __shared__ float red[8 * 256];

    const int tid  = threadIdx.x;
    const int w    = tid >> 5;
    const int lane = tid & 31;
    const int b    = blockIdx.x / DTILES;
    const int dt   = blockIdx.x % DTILES;

    const int m   = lane & 15;        // A-row (segment) / B-col index
    const int hi  = lane >> 4;        // K half select
    const int col = dt * 16 + m;

    const int KPW  = NPATCH / 8;      // 392 patches per wave
    const int kbeg = w * KPW;

    const float* ebase = emb + (size_t)b * NPATCH * DIM + col;
    const int*   sbase = seg_ids + b * NPATCH;

    v8f c0 = {0.f, 0.f, 0.f, 0.f, 0.f, 0.f, 0.f, 0.f};
    v8f c1 = {0.f, 0.f, 0.f, 0.f, 0.f, 0.f, 0.f, 0.f};

    for (int k0 = kbeg; k0 < kbeg + KPW; k0 += 8) {
        {   // K chunk [k0, k0+4)
            const float* pe = ebase + (size_t)(k0 + 2 * hi) * DIM;
            __builtin_prefetch(pe + 8 * DIM, 0, 0);          // next iteration
            int2 id = *(const int2*)(sbase + k0 + 2 * hi);   // broadcast-ish load
            v2f a;  a.x = (id.x == m) ? 1.f : 0.f;  a.y = (id.y == m) ? 1.f : 0.f;
            v2f bf; bf.x = pe[0];                   bf.y = pe[DIM];
            c0 = __builtin_amdgcn_wmma_f32_16x16x4_f32(
                     false, a, false, bf, (short)0, c0, false, false);
        }
        {   // K chunk [k0+4, k0+8) -> independent accumulator (breaks RAW chain)
            const int k1 = k0 + 4;
            const float* pe = ebase + (size_t)(k1 + 2 * hi) * DIM;
            int2 id = *(const int2*)(sbase + k1 + 2 * hi);
            v2f a;  a.x = (id.x == m) ? 1.f : 0.f;  a.y = (id.y == m) ? 1.f : 0.f;
            v2f bf; bf.x = pe[0];                   bf.y = pe[DIM];
            c1 = __builtin_amdgcn_wmma_f32_16x16x4_f32(
                     false, a, false, bf, (short)0, c1, false, false);
        }
    }

#pragma unroll
    for (int v = 0; v < 8; ++v) red[w * 256 + v * 32 + lane] = c0[v] + c1[v];
    __syncthreads();

    // Cross-wave reduce + mean + store: thread t owns (v = t/32, lane = t%32)
    const int fv = tid >> 5;
    const int fl = tid & 31;
    float s = 0.f;
#pragma unroll
    for (int ww = 0; ww < 8; ++ww) s += red[ww * 256 + fv * 32 + fl];
    const int fm = fv + (fl >> 4) * 8;     // segment index
    const int fn = fl & 15;                // column within tile
    const float cf  = (float)cnt[b * RSEG + fm];
    const float inv = 1.0f / fmaxf(cf, 1.0f);
    out[((size_t)b * RSEG + fm) * DIM + dt * 16 + fn] = s * inv;
}

// ---------------- host launch ----------------
extern "C" void kernel_launch(void* const* d_in, const int* in_sizes, int n_in,
                              void* d_out, int out_size, void* d_ws, size_t ws_size,
                              hipStream_t stream) {
    const float* emb = (const float*)d_in[0];   // [B, N, D] f32
    const int*   seg = (const int*)d_in[1];     // [B, IMG, IMG] i32
    (void)in_sizes; (void)n_in; (void)out_size; (void)ws_size;

    int* cnt     = (int*)d_ws;                  // [B*R] counts
    int* seg_ids = cnt + BATCH * RSEG;          // [B*N] dominant segment ids
    float* out   = (float*)d_out;               // [B, R, D]

    sppp_zero_counts<<<1, 512, 0, stream>>>(cnt);

    const int nblk1 = (BATCH * NPATCH + 255) / 256;
    sppp_seg_mode<<<nblk1, 256, 0, stream>>>(seg, seg_ids, cnt);

    sppp_pool_wmma<<<BATCH * DTILES, 256, 0, stream>>>(emb, seg_ids, cnt, out);
}